// DeformConv3D_19791209299987
// MI455X (gfx1250) — compile-verified
//
#include <hip/hip_runtime.h>
#include <math.h>

typedef float v2f __attribute__((ext_vector_type(2)));
typedef float v4f __attribute__((ext_vector_type(4)));
typedef float v8f __attribute__((ext_vector_type(8)));

#define G_    4
#define K2_   9
#define COFF  72            // 2*G*K2 offset channels
#define CIN   128
#define CG    32            // CIN / G
#define COUT  128
#define OG    32            // COUT / G
#define DD    6
#define BB    2
#define HIMG  56
#define WIMG  56
#define HW    3136
#define NIMG  12            // B*D
#define KOFF  (CIN * K2_)   // 1152 (offset-conv GEMM K)
#define KDEF  (CG * K2_)    // 288  (deform-conv GEMM K)
#define ASTR  292           // LDS row stride (floats): 16B aligned, conflict-free

static __device__ __forceinline__ v8f wmma_f32(v2f a, v2f b, v8f c) {
    return __builtin_amdgcn_wmma_f32_16x16x4_f32(false, a, false, b, (short)0, c,
                                                 false, false);
}

// ---------------------------------------------------------------------------
// Prep A: x (B,C,D,H,W) -> xt (n, hw, c) channels-last, LDS-tiled transpose
// ---------------------------------------------------------------------------
__global__ __launch_bounds__(256)
void dc_transpose_x(const float* __restrict__ x, float* __restrict__ xt) {
    __shared__ float tile[CIN][65];
    const int n = blockIdx.y, hw0 = blockIdx.x * 64;
    const int b = n / DD, d = n % DD;
    const int tid = threadIdx.x;
    for (int i = tid; i < CIN * 64; i += 256) {            // coalesced read
        const int c = i >> 6, p = i & 63;
        tile[c][p] = x[((size_t)(b * CIN + c) * DD + d) * HW + hw0 + p];
    }
    __syncthreads();
    for (int i = tid; i < CIN * 64; i += 256) {            // coalesced write
        const int p = i >> 7, c = i & 127;
        xt[((size_t)n * HW + hw0 + p) * CIN + c] = tile[c][p];
    }
}

// ---------------------------------------------------------------------------
// Prep B: weights -> K-contiguous layout: owt[oc][k2*128+c], cwt[oc][k2*32+c]
// ---------------------------------------------------------------------------
__global__ __launch_bounds__(256)
void dc_transpose_w(const float* __restrict__ ow, const float* __restrict__ cw,
                    float* __restrict__ owt, float* __restrict__ cwt) {
    const int i = blockIdx.x * 256 + threadIdx.x;
    if (i < COFF * KOFF) {
        const int oc = i / KOFF, kk = i % KOFF, c = kk / 9, k2 = kk % 9;
        owt[(size_t)oc * KOFF + k2 * CIN + c] = ow[i];
    }
    if (i < COUT * KDEF) {
        const int oc = i / KDEF, kk = i % KDEF, c = kk / 9, k2 = kk % 9;
        cwt[(size_t)oc * KDEF + k2 * CG + c] = cw[i];
    }
}

// ---------------------------------------------------------------------------
// Kernel 1: offset conv as implicit GEMM, K ordered k2-major (k2*128+c).
// One wave: 32 pixels (2 M-subtiles) x 80 outputs (5 N-tiles) = 10 accums.
// Hot loop: 2 b64 A loads + 5 b64 B loads + 10 WMMA, no div/mod.
// ---------------------------------------------------------------------------
__global__ __launch_bounds__(32)
void dc_offset_conv(const float* __restrict__ xt, const float* __restrict__ owt,
                    const float* __restrict__ ob, float* __restrict__ off_out) {
    const int lane = threadIdx.x;
    const int mrow = lane & 15;          // A row / B col / D col
    const int pair = lane >> 4;          // K-pair select (A/B), M-half (D)
    const int m0   = blockIdx.x * 32;    // 3136 % 32 == 0: tile within one n
    const int n    = m0 / HW;
    const int hwb  = m0 % HW;

    int hA[2], wA[2];
#pragma unroll
    for (int s = 0; s < 2; ++s) {
        const int hw = hwb + s * 16 + mrow;
        hA[s] = hw / WIMG; wA[s] = hw % WIMG;
    }
    const float* xn = xt + (size_t)n * HW * CIN;
    const float* owp[5];
#pragma unroll
    for (int t = 0; t < 5; ++t) {        // clamp padded oc rows (never stored)
        int oc = t * 16 + mrow; if (oc > COFF - 1) oc = COFF - 1;
        owp[t] = owt + (size_t)oc * KOFF;
    }

    v8f acc[2][5];
#pragma unroll
    for (int s = 0; s < 2; ++s)
#pragma unroll
        for (int t = 0; t < 5; ++t)
            acc[s][t] = v8f{0.f,0.f,0.f,0.f,0.f,0.f,0.f,0.f};

    for (int k2 = 0; k2 < K2_; ++k2) {
        const int ky = k2 / 3 - 1, kx = k2 % 3 - 1;
        const float* xs[2]; float aval[2];
#pragma unroll
        for (int s = 0; s < 2; ++s) {
            const int hy = hA[s] + ky, wx = wA[s] + kx;
            aval[s] = (hy >= 0 && hy < HIMG && wx >= 0 && wx < WIMG) ? 1.f : 0.f;
            const int hyc = min(max(hy, 0), HIMG - 1);
            const int wxc = min(max(wx, 0), WIMG - 1);
            xs[s] = xn + (size_t)(hyc * WIMG + wxc) * CIN;
        }
        const int kbase = k2 * CIN;
        for (int kb = 0; kb < CIN / 4; ++kb) {
            const int cb = kb * 4 + pair * 2;
            v2f a0 = *(const v2f*)(xs[0] + cb); a0 *= aval[0];
            v2f a1 = *(const v2f*)(xs[1] + cb); a1 *= aval[1];
#pragma unroll
            for (int t = 0; t < 5; ++t) {
                const v2f bv = *(const v2f*)(owp[t] + kbase + cb);
                acc[0][t] = wmma_f32(a0, bv, acc[0][t]);
                acc[1][t] = wmma_f32(a1, bv, acc[1][t]);
            }
        }
    }
    // D layout: VGPR r of lane l = D[8*(l>>4)+r][l&15]
#pragma unroll
    for (int t = 0; t < 5; ++t) {
        const int oc = t * 16 + mrow;
        if (oc < COFF) {
            const float bias = ob[oc];
#pragma unroll
            for (int s = 0; s < 2; ++s)
#pragma unroll
                for (int r = 0; r < 8; ++r) {
                    const int hw2 = hwb + s * 16 + pair * 8 + r;
                    off_out[((size_t)n * COFF + oc) * HW + hw2] = acc[s][t][r] + bias;
                }
        }
    }
}

// ---------------------------------------------------------------------------
// Kernel 2: deformable conv. Grid (196, G, NIMG), one wave each.
// Bilinear gather with b128 channels-last corner loads -> LDS (b128 stores),
// then 72 K-steps x 2 fp32 WMMA. Validity folded into bilinear weights.
// ---------------------------------------------------------------------------
__global__ __launch_bounds__(32)
void dc_deform_conv(const float* __restrict__ xt, const float* __restrict__ off,
                    const float* __restrict__ cwt, const float* __restrict__ cb,
                    float* __restrict__ y) {
    __shared__ float As[16 * ASTR];

    const int mtile = blockIdx.x, g = blockIdx.y, n = blockIdx.z;
    const int lane = threadIdx.x;
    const int p    = lane & 15;          // pixel (A row / B col)
    const int half = lane >> 4;          // channel half (stage), K pair (wmma)
    const int hw0  = mtile * 16, hw = hw0 + p;
    const int h    = hw / WIMG, w = hw % WIMG;

    const float* xn = xt + (size_t)n * HW * CIN + g * CG + half * 16;

    for (int k2 = 0; k2 < K2_; ++k2) {
        const int ky = k2 / 3 - 1, kx = k2 % 3 - 1;
        const float oy = off[((size_t)n * COFF + (g * K2_ + k2) * 2 + 0) * HW + hw];
        const float ox = off[((size_t)n * COFF + (g * K2_ + k2) * 2 + 1) * HW + hw];
        const float py = oy + (float)(h + ky);
        const float px = ox + (float)(w + kx);
        const float y0f = floorf(py), x0f = floorf(px);
        const float ly = py - y0f, lx = px - x0f;
        const int y0 = (int)y0f, x0 = (int)x0f;
        const int y1 = y0 + 1,   x1 = x0 + 1;
        const float vy0 = (y0 >= 0 && y0 < HIMG) ? 1.f : 0.f;
        const float vy1 = (y1 >= 0 && y1 < HIMG) ? 1.f : 0.f;
        const float vx0 = (x0 >= 0 && x0 < WIMG) ? 1.f : 0.f;
        const float vx1 = (x1 >= 0 && x1 < WIMG) ? 1.f : 0.f;
        const int yc0 = min(max(y0, 0), HIMG - 1);
        const int yc1 = min(max(y1, 0), HIMG - 1);
        const int xc0 = min(max(x0, 0), WIMG - 1);
        const int xc1 = min(max(x1, 0), WIMG - 1);
        const float w00 = (1.f - ly) * (1.f - lx) * (vy0 * vx0);
        const float w01 = (1.f - ly) * lx         * (vy0 * vx1);
        const float w10 = ly * (1.f - lx)         * (vy1 * vx0);
        const float w11 = ly * lx                 * (vy1 * vx1);
        const float* c00 = xn + (size_t)(yc0 * WIMG + xc0) * CIN;
        const float* c01 = xn + (size_t)(yc0 * WIMG + xc1) * CIN;
        const float* c10 = xn + (size_t)(yc1 * WIMG + xc0) * CIN;
        const float* c11 = xn + (size_t)(yc1 * WIMG + xc1) * CIN;
        float* dst = &As[p * ASTR + k2 * CG + half * 16];
#pragma unroll
        for (int q = 0; q < 4; ++q) {    // 4 channels per b128
            const v4f s = w00 * (*(const v4f*)(c00 + q * 4))
                        + w01 * (*(const v4f*)(c01 + q * 4))
                        + w10 * (*(const v4f*)(c10 + q * 4))
                        + w11 * (*(const v4f*)(c11 + q * 4));
            *(v4f*)(dst + q * 4) = s;
        }
    }
    __syncthreads();

    // GEMM: out[p][o] = sum_k As[p][k] * cwt[g*32+o][k], kidx = k2*32+c
    v8f acc0 = v8f{0.f,0.f,0.f,0.f,0.f,0.f,0.f,0.f};
    v8f acc1 = v8f{0.f,0.f,0.f,0.f,0.f,0.f,0.f,0.f};
    const int ocb = g * OG;
    const float* b0p = cwt + (size_t)(ocb + p) * KDEF;
    const float* b1p = cwt + (size_t)(ocb + 16 + p) * KDEF;
    const float* ap  = &As[p * ASTR + half * 2];
    for (int kb = 0; kb < KDEF / 4; ++kb) {
        const v2f a  = *(const v2f*)(ap + kb * 4);
        const v2f b0 = *(const v2f*)(b0p + kb * 4 + half * 2);
        const v2f b1 = *(const v2f*)(b1p + kb * 4 + half * 2);
        acc0 = wmma_f32(a, b0, acc0);
        acc1 = wmma_f32(a, b1, acc1);
    }
    const float bias0 = cb[ocb + p];
    const float bias1 = cb[ocb + 16 + p];
#pragma unroll
    for (int r = 0; r < 8; ++r) {
        const int m = hw0 + half * 8 + r;
        y[((size_t)n * COUT + (ocb + p)) * HW + m]      = acc0[r] + bias0;
        y[((size_t)n * COUT + (ocb + 16 + p)) * HW + m] = acc1[r] + bias1;
    }
}

// ---------------------------------------------------------------------------
// Kernel 3: per-(b,cout) instance norm over (D,H,W) + exact GELU.
// ---------------------------------------------------------------------------
__global__ __launch_bounds__(256)
void dc_norm_gelu(const float* __restrict__ y, float* __restrict__ out) {
    __shared__ float s_sum[256];
    __shared__ float s_sq[256];
    const int co  = blockIdx.x & 127;
    const int b   = blockIdx.x >> 7;
    const int tid = threadIdx.x;
    const int total = DD * HW;

    float sum = 0.f, sq = 0.f;
    for (int i = tid; i < total; i += 256) {
        const int d = i / HW, hw = i % HW;
        const float v = y[((size_t)(b * DD + d) * COUT + co) * HW + hw];
        sum += v; sq += v * v;
    }
    s_sum[tid] = sum; s_sq[tid] = sq;
    __syncthreads();
    for (int s = 128; s > 0; s >>= 1) {
        if (tid < s) { s_sum[tid] += s_sum[tid + s]; s_sq[tid] += s_sq[tid + s]; }
        __syncthreads();
    }
    const float mu  = s_sum[0] / (float)total;
    const float var = s_sq[0] / (float)total - mu * mu;
    const float inv = rsqrtf(var + 1e-5f);

    for (int i = tid; i < total; i += 256) {
        const int d = i / HW, hw = i % HW;
        const float v =
            (y[((size_t)(b * DD + d) * COUT + co) * HW + hw] - mu) * inv;
        const float gl = 0.5f * v * (1.f + erff(v * 0.7071067811865476f));
        out[((size_t)(b * COUT + co) * DD + d) * HW + hw] = gl;
    }
}

// ---------------------------------------------------------------------------
extern "C" void kernel_launch(void* const* d_in, const int* in_sizes, int n_in,
                              void* d_out, int out_size, void* d_ws, size_t ws_size,
                              hipStream_t stream) {
    const float* x  = (const float*)d_in[0];   // (2,128,6,56,56)
    const float* ow = (const float*)d_in[1];   // (72,128,3,3)
    const float* ob = (const float*)d_in[2];   // (72,)
    const float* cw = (const float*)d_in[3];   // (128,32,3,3)
    const float* cb = (const float*)d_in[4];   // (128,)
    float* out = (float*)d_out;

    float* xt   = (float*)d_ws;                         // 12*3136*128
    float* owt  = xt   + (size_t)NIMG * HW * CIN;       // 72*1152
    float* cwt  = owt  + (size_t)COFF * KOFF;           // 128*288
    float* offo = cwt  + (size_t)COUT * KDEF;           // 12*72*3136
    float* y    = offo + (size_t)NIMG * COFF * HW;      // 12*128*3136

    dc_transpose_x<<<dim3(HW / 64, NIMG), dim3(256), 0, stream>>>(x, xt);
    dc_transpose_w<<<dim3((COFF * KOFF + 255) / 256), dim3(256), 0, stream>>>(
        ow, cw, owt, cwt);
    dc_offset_conv<<<dim3((NIMG * HW) / 32), dim3(32), 0, stream>>>(
        xt, owt, ob, offo);
    dc_deform_conv<<<dim3(HW / 16, G_, NIMG), dim3(32), 0, stream>>>(
        xt, offo, cwt, cb, y);
    dc_norm_gelu<<<dim3(BB * COUT), dim3(256), 0, stream>>>(y, out);
}